// Histogram_Binning_79070347919528
// MI455X (gfx1250) — compile-verified
//
#include <hip/hip_runtime.h>
#include <hip/hip_bf16.h>

typedef __attribute__((ext_vector_type(4))) float f32x4;
typedef __attribute__((ext_vector_type(4))) int   v4i;

#if defined(__gfx1250__)
typedef __attribute__((address_space(1))) v4i* as1_v4i_p;  // global ("__device__")
typedef __attribute__((address_space(3))) v4i* as3_v4i_p;  // LDS ("__shared__")
#endif

namespace {
constexpr int kC    = 8;
constexpr int kBins = 15;
constexpr int kHW   = 512 * 512;
constexpr int kGroupsPerImage = kHW / 4;   // 65536 float4-groups per (b) image
constexpr int kLut  = kC * kBins;          // 120 floats = 480 bytes
}

__global__ __launch_bounds__(256) void cal_hist_kernel(
    const float* __restrict__ logits,
    const float* __restrict__ val_freqs,
    float* __restrict__ out,
    int n_groups)
{
  __shared__ float s_lut[kLut];

#if defined(__gfx1250__)
  // CDNA5 async DMA: 30 lanes x B128 = 480 B of LUT, global -> LDS, no VGPR round trip.
  if (threadIdx.x < kLut / 4) {
    float* src = const_cast<float*>(val_freqs) + threadIdx.x * 4;
#if __has_builtin(__builtin_amdgcn_global_load_async_to_lds_b128)
    __builtin_amdgcn_global_load_async_to_lds_b128(
        (as1_v4i_p)(v4i*)src,
        (as3_v4i_p)(v4i*)&s_lut[threadIdx.x * 4],
        0, 0);
#else
    unsigned lds_off = threadIdx.x * 16u;  // s_lut is the only LDS object -> base offset 0
    asm volatile("global_load_async_to_lds_b128 %0, %1, off"
                 :: "v"(lds_off), "v"(src)
                 : "memory");
#endif
  }
#if __has_builtin(__builtin_amdgcn_s_wait_asynccnt)
  __builtin_amdgcn_s_wait_asynccnt(0);
#else
  asm volatile("s_wait_asynccnt 0x0" ::: "memory");
#endif
#else
  // Host-pass / non-gfx1250 semantic fallback.
  for (int i = threadIdx.x; i < kLut; i += blockDim.x) s_lut[i] = val_freqs[i];
#endif
  __syncthreads();

  const int stride = (int)(gridDim.x * blockDim.x);
  for (int g = (int)(blockIdx.x * blockDim.x + threadIdx.x); g < n_groups; g += stride) {
    const int b = g >> 16;                       // g / kGroupsPerImage
    const int r = g & (kGroupsPerImage - 1);     // g % kGroupsPerImage
    const size_t off = (size_t)b * kC * kHW + (size_t)r * 4;
    const float* base = logits + off;

    // Prefetch next grid-stride iteration's 8 class planes (global_prefetch_b8).
    const int gn = g + stride;
    if (gn < n_groups) {
      const int bn = gn >> 16;
      const int rn = gn & (kGroupsPerImage - 1);
      const float* nb = logits + (size_t)bn * kC * kHW + (size_t)rn * 4;
#pragma unroll
      for (int c = 0; c < kC; ++c) __builtin_prefetch(nb + (size_t)c * kHW, 0, 1);
    }

    // 8 coalesced B128 non-temporal loads (streaming: touched once, > L2 footprint).
    f32x4 v[kC];
#pragma unroll
    for (int c = 0; c < kC; ++c)
      v[c] = __builtin_nontemporal_load((const f32x4*)(base + (size_t)c * kHW));

#pragma unroll
    for (int j = 0; j < 4; ++j) {
      // softmax over class dim (max-subtracted, like jax.nn.softmax)
      float m = v[0][j];
#pragma unroll
      for (int c = 1; c < kC; ++c) m = fmaxf(m, v[c][j]);
      float e[kC];
      float s = 0.f;
#pragma unroll
      for (int c = 0; c < kC; ++c) { e[c] = __expf(v[c][j] - m); s += e[c]; }
      const float inv = 1.0f / s;

      // bin, LUT gather, renormalize
      float cal[kC];
      float cs = 0.f;
#pragma unroll
      for (int c = 0; c < kC; ++c) {
        const float p = e[c] * inv;              // p in [0,1]
        int bin = (int)(p * (float)kBins);       // trunc == floor for p >= 0
        bin = bin < 0 ? 0 : (bin > kBins - 1 ? kBins - 1 : bin);
        const float f = s_lut[c * kBins + bin];
        cal[c] = f;
        cs += f;
      }
      const float invc = 1.0f / cs;
#pragma unroll
      for (int c = 0; c < kC; ++c) v[c][j] = cal[c] * invc;  // reuse v[] as output regs
    }

    // 8 coalesced B128 non-temporal stores.
    float* ob = out + off;
#pragma unroll
    for (int c = 0; c < kC; ++c)
      __builtin_nontemporal_store(v[c], (f32x4*)(ob + (size_t)c * kHW));
  }
}

extern "C" void kernel_launch(void* const* d_in, const int* in_sizes, int n_in,
                              void* d_out, int out_size, void* d_ws, size_t ws_size,
                              hipStream_t stream) {
  const float* logits    = (const float*)d_in[0];
  const float* val_freqs = (const float*)d_in[1];
  float* out = (float*)d_out;

  const int total    = in_sizes[0];          // B*C*H*W = 33,554,432
  const int n_groups = total / (kC * 4);     // 1,048,576 float4-groups
  const int threads  = 256;                  // 8 wave32 waves per block
  const int blocks   = 1024;                 // 4 grid-stride iters/thread -> prefetch useful

  hipLaunchKernelGGL(cal_hist_kernel, dim3(blocks), dim3(threads), 0, stream,
                     logits, val_freqs, out, n_groups);
}